// GNN_51539607552553
// MI455X (gfx1250) — compile-verified
//
#include <hip/hip_runtime.h>

// ---------------------------------------------------------------------------
// Problem constants (from reference): B=2, H=W=192, HID=10
// ---------------------------------------------------------------------------
#define HWL   36864L          // 192*192
#define NPOS  73728L          // B*H*W  (BN reduction size, GEMM N dimension)
#define EPSV  1e-5f

#define KC    32              // K-chunk staged in LDS per pipeline stage
#define XPAD  144             // row stride (floats) of X tile in LDS (bank spread)
#define WPAD  34              // row stride (floats) of W tile in LDS (even -> b64 ok)

typedef float v2f __attribute__((ext_vector_type(2)));
typedef float v4f __attribute__((ext_vector_type(4)));
typedef float v8f __attribute__((ext_vector_type(8)));

// generic->LDS: on gfx1250 the LDS aperture keeps the LDS byte offset in
// addr[31:0] (ISA 10.2), so truncating a generic pointer gives the DS address.
__device__ __forceinline__ unsigned lds_addr32(const void* p) {
    return (unsigned)(unsigned long long)p;
}

// ---------------------------------------------------------------------------
// GEMM  y[co, n] = sum_k W[co,k] * X[k, n]    (X = virtual concat of <=3 segs)
// plus per-channel sum / sumsq for batch-norm.
// 256 threads = 8 waves; block owns 128 columns; wave owns a 16-col tile and
// all MT m-tiles (MT = ceil(co/16), compile-time).  X staged to LDS with
// async-to-LDS b128 loads, double buffered; W chunk staged zero-padded;
// inner loop is branch-free ds_load + V_WMMA_F32_16X16X4_F32.
// ---------------------------------------------------------------------------
template<int MT>
__global__ __launch_bounds__(256) void gemm_bnstats(
    const float* __restrict__ s0p, int s0c, long s0bs,
    const float* __restrict__ s1p, int s1c, long s1bs,
    const float* __restrict__ s2p, int s2c, long s2bs,
    const float* __restrict__ W, int ci, int co,
    float* __restrict__ y, float* __restrict__ stats)
{
    __shared__ float sX[2][KC * XPAD];   // 2 x 18,432 B
    __shared__ float sW[2][80 * WPAD];   // 2 x 10,880 B
    __shared__ float sred[160];          // [0,80): sum  [80,160): sumsq

    const int tid  = threadIdx.x;
    const int wave = tid >> 5;
    const int lane = tid & 31;
    const int l16  = lane & 15;
    const int hi   = lane >> 4;

    for (int i = tid; i < 160; i += 256) sred[i] = 0.0f;

    const long n0  = (long)blockIdx.x * 128;   // HW%128==0 -> block never straddles b
    const int  b   = (int)(n0 / HWL);
    const long hw0 = n0 % HWL;

    const float* base0 = s0p + (long)b * s0bs;
    const float* base1 = s1p ? (s1p + (long)b * s1bs) : nullptr;
    const float* base2 = s2p ? (s2p + (long)b * s2bs) : nullptr;
    const int c01 = s0c + s1c;

    const int nchunks = (ci + KC - 1) / KC;

    // ---- stage one K-chunk of X into LDS buffer `buf` (async) --------------
    auto stageX = [&](int buf, int k0) {
        for (int r = wave; r < KC; r += 8) {            // one row per wave pass
            const int k = k0 + r;
            float* dstp = &sX[buf][r * XPAD + lane * 4];
            if (k < ci) {
                const float* bp; int c;
                if (k < s0c)      { bp = base0; c = k; }
                else if (k < c01) { bp = base1; c = k - s0c; }
                else              { bp = base2; c = k - c01; }
                const float* src = bp + (long)c * HWL + hw0 + lane * 4;
                asm volatile("global_load_async_to_lds_b128 %0, %1, off"
                             :: "v"(lds_addr32(dstp)),
                                "v"((unsigned long long)src)
                             : "memory");
            } else {
                *(v4f*)dstp = (v4f){0.0f, 0.0f, 0.0f, 0.0f};
            }
        }
    };
    // ---- stage one K-chunk of W (zero padded) into LDS ---------------------
    auto stageW = [&](int buf, int k0) {
        for (int idx = tid; idx < 80 * KC; idx += 256) {
            const int row = idx >> 5, kk = idx & 31, k = k0 + kk;
            float v = 0.0f;
            if (row < co && k < ci) v = W[(long)row * ci + k];
            sW[buf][row * WPAD + kk] = v;
        }
    };

    stageX(0, 0);
    stageW(0, 0);
    asm volatile("s_wait_asynccnt 0x0" ::: "memory");
    __syncthreads();

    v8f acc[MT] = {};
    const int colLocal = (wave << 4) + l16;            // 0..127

    for (int ch = 0; ch < nchunks; ++ch) {
        const int cur = ch & 1;
        if (ch + 1 < nchunks) {                        // prefetch next chunk
            stageX(cur ^ 1, (ch + 1) * KC);
            stageW(cur ^ 1, (ch + 1) * KC);
        }
        const float* sx = &sX[cur][0];
        const float* sw = &sW[cur][0];
#pragma unroll
        for (int kt = 0; kt < KC / 4; ++kt) {
            const int ka = (kt << 2) + (hi << 1);      // this lane's K pair
            // B tile (4x16): lane = column, lane-half = K group
            v2f bv;
            bv.x = sx[ka * XPAD + colLocal];
            bv.y = sx[(ka + 1) * XPAD + colLocal];
            // A tiles (16x4): lane = row M, lane-half = K group
            v2f av[MT];
#pragma unroll
            for (int m = 0; m < MT; ++m)
                av[m] = *(const v2f*)&sw[((m << 4) + l16) * WPAD + ka];
#pragma unroll
            for (int m = 0; m < MT; ++m)
                acc[m] = __builtin_amdgcn_wmma_f32_16x16x4_f32(
                    false, av[m], false, bv, (short)0, acc[m], false, false);
        }
        asm volatile("s_wait_asynccnt 0x0" ::: "memory"); // next chunk landed
        __syncthreads();                                  // all waves done w/ cur
    }

    // ---- write D and reduce per-channel sums -------------------------------
    const long col = hw0 + colLocal;
#pragma unroll
    for (int m = 0; m < MT; ++m) {
#pragma unroll
        for (int j = 0; j < 8; ++j) {
            const int row = (m << 4) + (hi << 3) + j;
            const float v = acc[m][j];
            float s = v, q = v * v;
            s += __shfl_xor(s, 1, 32);  q += __shfl_xor(q, 1, 32);
            s += __shfl_xor(s, 2, 32);  q += __shfl_xor(q, 2, 32);
            s += __shfl_xor(s, 4, 32);  q += __shfl_xor(q, 4, 32);
            s += __shfl_xor(s, 8, 32);  q += __shfl_xor(q, 8, 32);
            if (row < co) {
                y[(long)b * co * HWL + (long)row * HWL + col] = v;
                if (l16 == 0) {
                    atomicAdd(&sred[row], s);
                    atomicAdd(&sred[80 + row], q);
                }
            }
        }
    }
    __syncthreads();
    for (int c = tid; c < co; c += 256) {
        atomicAdd(&stats[c],      sred[c]);
        atomicAdd(&stats[co + c], sred[80 + c]);
    }
}

// ---------------------------------------------------------------------------
// BN finalize: stats -> (scale, shift) per channel.
// ---------------------------------------------------------------------------
__global__ void bn_finalize(const float* __restrict__ stats,
                            const float* __restrict__ bn,   // [gamma[co], beta[co]]
                            float* __restrict__ ss, int co)
{
    const int c = threadIdx.x;
    if (c < co) {
        const float inv  = 1.0f / (float)NPOS;
        const float mean = stats[c] * inv;
        const float var  = stats[co + c] * inv - mean * mean;
        const float scale = bn[c] * rsqrtf(var + EPSV);
        ss[c]      = scale;
        ss[co + c] = bn[co + c] - mean * scale;
    }
}

// ---------------------------------------------------------------------------
// In-place  y = relu(y*scale + shift) [+ a1] [+ a2]
// ---------------------------------------------------------------------------
__global__ void cbr_add_inplace(float* __restrict__ y, const float* __restrict__ ss, int C,
                                const float* __restrict__ a1, long a1bs, int a1off,
                                const float* __restrict__ a2, long a2bs, int a2off)
{
    const long idx = (long)blockIdx.x * 256 + threadIdx.x;
    const long total = 2L * C * HWL;
    if (idx >= total) return;
    const long hw = idx % HWL;
    const int  c  = (int)((idx / HWL) % C);
    const long b  = idx / ((long)C * HWL);
    float v = fmaxf(fmaf(y[idx], ss[c], ss[C + c]), 0.0f);
    if (a1) v += a1[b * a1bs + (long)(a1off + c) * HWL + hw];
    if (a2) v += a2[b * a2bs + (long)(a2off + c) * HWL + hw];
    y[idx] = v;
}

// xp_new (B,60,HW) = xp[:, slices 1..6] + xhp[:, ch 10..69]
__global__ void xpnew_kernel(float* __restrict__ dst, const float* __restrict__ xp,
                             const float* __restrict__ xhp)
{
    const long idx = (long)blockIdx.x * 256 + threadIdx.x;
    const long total = 2L * 60 * HWL;
    if (idx >= total) return;
    const long hw = idx % HWL;
    const int  c  = (int)((idx / HWL) % 60);
    const long b  = idx / (60L * HWL);
    const long o  = b * 70 * HWL + (long)(10 + c) * HWL + hw;
    dst[idx] = xp[o] + xhp[o];
}

// out (B,220,HW) = concat[cbr(y_pu), cbr(y_hu), cbr(y_fu), xfh, xhp]
__global__ void assemble_out(float* __restrict__ out,
                             const float* __restrict__ y_pu, const float* __restrict__ ss_pu,
                             const float* __restrict__ y_hu, const float* __restrict__ ss_hu,
                             const float* __restrict__ y_fu, const float* __restrict__ ss_fu,
                             const float* __restrict__ xfh,  const float* __restrict__ xhp)
{
    const long idx = (long)blockIdx.x * 256 + threadIdx.x;
    const long total = 2L * 220 * HWL;
    if (idx >= total) return;
    const long hw = idx % HWL;
    const int  c  = (int)((idx / HWL) % 220);
    const long b  = idx / (220L * HWL);
    float v;
    if (c < 70) {
        const float t = y_pu[b * 70 * HWL + (long)c * HWL + hw];
        v = fmaxf(fmaf(t, ss_pu[c], ss_pu[70 + c]), 0.0f);
    } else if (c < 100) {
        const int cc = c - 70;
        const float t = y_hu[b * 30 * HWL + (long)cc * HWL + hw];
        v = fmaxf(fmaf(t, ss_hu[cc], ss_hu[30 + cc]), 0.0f);
    } else if (c < 120) {
        const int cc = c - 100;
        const float t = y_fu[b * 20 * HWL + (long)cc * HWL + hw];
        v = fmaxf(fmaf(t, ss_fu[cc], ss_fu[20 + cc]), 0.0f);
    } else if (c < 150) {
        v = xfh[b * 30 * HWL + (long)(c - 120) * HWL + hw];
    } else {
        v = xhp[b * 70 * HWL + (long)(c - 150) * HWL + hw];
    }
    out[idx] = v;
}

// ---------------------------------------------------------------------------
// Host orchestration
// ---------------------------------------------------------------------------
extern "C" void kernel_launch(void* const* d_in, const int* in_sizes, int n_in,
                              void* d_out, int out_size, void* d_ws, size_t ws_size,
                              hipStream_t stream)
{
    (void)in_sizes; (void)n_in; (void)out_size;
    const float* xp    = (const float*)d_in[0];
    const float* xh    = (const float*)d_in[1];
    const float* xf    = (const float*)d_in[2];
    const float* p_fea = (const float*)d_in[3];
    const float* h_fea = (const float*)d_in[4];
    const float* f_fea = (const float*)d_in[5];
    const float* w_hf1  = (const float*)d_in[6];  const float* bn_hf1  = (const float*)d_in[7];
    const float* w_hf2  = (const float*)d_in[8];  const float* bn_hf2  = (const float*)d_in[9];
    const float* w_phu1 = (const float*)d_in[10]; const float* bn_phu1 = (const float*)d_in[11];
    const float* w_phu2 = (const float*)d_in[12]; const float* bn_phu2 = (const float*)d_in[13];
    const float* w_phl1 = (const float*)d_in[14]; const float* bn_phl1 = (const float*)d_in[15];
    const float* w_phl2 = (const float*)d_in[16]; const float* bn_phl2 = (const float*)d_in[17];
    const float* w_fd1  = (const float*)d_in[18]; const float* bn_fd1  = (const float*)d_in[19];
    const float* w_fd2  = (const float*)d_in[20]; const float* bn_fd2  = (const float*)d_in[21];
    const float* w_hd1  = (const float*)d_in[22]; const float* bn_hd1  = (const float*)d_in[23];
    const float* w_hd2  = (const float*)d_in[24]; const float* bn_hd2  = (const float*)d_in[25];
    const float* w_fu   = (const float*)d_in[26]; const float* bn_fu   = (const float*)d_in[27];
    const float* w_hu   = (const float*)d_in[28]; const float* bn_hu   = (const float*)d_in[29];
    const float* w_pu   = (const float*)d_in[30]; const float* bn_pu   = (const float*)d_in[31];

    float* ws = (float*)d_ws;
    long off = 0;
    auto alloc = [&](long n) { float* p = ws + off; off += n; return p; };
    float* y_hf1  = alloc(2L * 20 * HWL);
    float* y_hf2  = alloc(2L * 10 * HWL);
    float* y_fu   = alloc(2L * 20 * HWL);
    float* y_fd1  = alloc(2L * 30 * HWL);
    float* y_fd2  = alloc(2L * 30 * HWL);   // becomes xfh in place
    float* y_phu1 = alloc(2L * 20 * HWL);
    float* y_phu2 = alloc(2L * 10 * HWL);   // becomes xhu
    float* y_phl1 = alloc(2L * 20 * HWL);
    float* y_phl2 = alloc(2L * 10 * HWL);   // becomes xhl
    float* y_hu   = alloc(2L * 30 * HWL);
    float* y_hd1  = alloc(2L * 70 * HWL);
    float* y_hd2  = alloc(2L * 70 * HWL);   // becomes xhp
    float* y_pu   = alloc(2L * 70 * HWL);
    float* xpnew  = alloc(2L * 60 * HWL);
    float* stats  = alloc(13L * 160);
    float* ssbuf  = alloc(13L * 160);
    if ((size_t)(off * sizeof(float)) > ws_size) return;  // workspace too small

    auto st = [&](int i) { return stats + i * 160; };
    auto sl = [&](int i) { return ssbuf + i * 160; };

    hipMemsetAsync(stats, 0, 13 * 160 * sizeof(float), stream);

    const dim3 GB(576), GT(256);   // 73728 cols / 128 per block
    auto gemm = [&](const float* s0p, int s0c, long s0bs,
                    const float* s1p, int s1c, long s1bs,
                    const float* s2p, int s2c, long s2bs,
                    const float* W, int ci, int co, float* y, float* s) {
        const int mt = (co + 15) >> 4;
        if (mt == 1)
            gemm_bnstats<1><<<GB, GT, 0, stream>>>(s0p, s0c, s0bs, s1p, s1c, s1bs,
                                                   s2p, s2c, s2bs, W, ci, co, y, s);
        else if (mt == 2)
            gemm_bnstats<2><<<GB, GT, 0, stream>>>(s0p, s0c, s0bs, s1p, s1c, s1bs,
                                                   s2p, s2c, s2bs, W, ci, co, y, s);
        else
            gemm_bnstats<5><<<GB, GT, 0, stream>>>(s0p, s0c, s0bs, s1p, s1c, s1bs,
                                                   s2p, s2c, s2bs, W, ci, co, y, s);
    };
    auto fin = [&](int i, const float* bn, int co) {
        bn_finalize<<<1, 128, 0, stream>>>(st(i), bn, sl(i), co);
    };
    auto act = [&](float* y, int i, int C,
                   const float* a1, long a1bs, int a1off,
                   const float* a2, long a2bs, int a2off) {
        const long total = 2L * C * HWL;
        cbr_add_inplace<<<(unsigned)((total + 255) / 256), 256, 0, stream>>>(
            y, sl(i), C, a1, a1bs, a1off, a2, a2bs, a2off);
    };

    // hf1: [xh1; xh2] (ci 20, co 20)
    gemm(xh + 10 * HWL, 10, 30 * HWL, xh + 20 * HWL, 10, 30 * HWL,
         nullptr, 0, 0, w_hf1, 20, 20, y_hf1, st(0));
    fin(0, bn_hf1, 20);
    act(y_hf1, 0, 20, nullptr, 0, 0, nullptr, 0, 0);
    // hf2 -> xf_new = cbr + xf[:,1]
    gemm(y_hf1, 20, 20 * HWL, nullptr, 0, 0, nullptr, 0, 0, w_hf2, 20, 10, y_hf2, st(1));
    fin(1, bn_hf2, 10);
    act(y_hf2, 1, 10, xf, 20 * HWL, 10, nullptr, 0, 0);
    // fu: [f_fea; xf_new] (266 -> 20); raw y kept, cbr applied at assembly
    gemm(f_fea, 256, 256 * HWL, y_hf2, 10, 10 * HWL, nullptr, 0, 0,
         w_fu, 266, 20, y_fu, st(10));
    fin(10, bn_fu, 20);
    // fd1: [h_fea; xf1] (266 -> 30)
    gemm(h_fea, 256, 256 * HWL, xf + 10 * HWL, 10, 20 * HWL, nullptr, 0, 0,
         w_fd1, 266, 30, y_fd1, st(6));
    fin(6, bn_fd1, 30);
    act(y_fd1, 6, 30, nullptr, 0, 0, nullptr, 0, 0);
    // fd2 -> xfh
    gemm(y_fd1, 30, 30 * HWL, nullptr, 0, 0, nullptr, 0, 0, w_fd2, 30, 30, y_fd2, st(7));
    fin(7, bn_fd2, 30);
    act(y_fd2, 7, 30, nullptr, 0, 0, nullptr, 0, 0);
    // phu1: xp[:,1:5] (40 -> 20)
    gemm(xp + 10 * HWL, 40, 70 * HWL, nullptr, 0, 0, nullptr, 0, 0,
         w_phu1, 40, 20, y_phu1, st(2));
    fin(2, bn_phu1, 20);
    act(y_phu1, 2, 20, nullptr, 0, 0, nullptr, 0, 0);
    // phu2 -> xhu = cbr + xh1 + xfh1
    gemm(y_phu1, 20, 20 * HWL, nullptr, 0, 0, nullptr, 0, 0, w_phu2, 20, 10, y_phu2, st(3));
    fin(3, bn_phu2, 10);
    act(y_phu2, 3, 10, xh, 30 * HWL, 10, y_fd2, 30 * HWL, 10);
    // phl1: xp[:,5:7] (20 -> 20)
    gemm(xp + 50 * HWL, 20, 70 * HWL, nullptr, 0, 0, nullptr, 0, 0,
         w_phl1, 20, 20, y_phl1, st(4));
    fin(4, bn_phl1, 20);
    act(y_phl1, 4, 20, nullptr, 0, 0, nullptr, 0, 0);
    // phl2 -> xhl = cbr + xh2 + xfh2
    gemm(y_phl1, 20, 20 * HWL, nullptr, 0, 0, nullptr, 0, 0, w_phl2, 20, 10, y_phl2, st(5));
    fin(5, bn_phl2, 10);
    act(y_phl2, 5, 10, xh, 30 * HWL, 20, y_fd2, 30 * HWL, 20);
    // hu: [h_fea; xhu; xhl] (276 -> 30)
    gemm(h_fea, 256, 256 * HWL, y_phu2, 10, 10 * HWL, y_phl2, 10, 10 * HWL,
         w_hu, 276, 30, y_hu, st(11));
    fin(11, bn_hu, 30);
    // hd1: [p_fea; xh1; xh2] (276 -> 70)
    gemm(p_fea, 256, 256 * HWL, xh + 10 * HWL, 10, 30 * HWL, xh + 20 * HWL, 10, 30 * HWL,
         w_hd1, 276, 70, y_hd1, st(8));
    fin(8, bn_hd1, 70);
    act(y_hd1, 8, 70, nullptr, 0, 0, nullptr, 0, 0);
    // hd2 -> xhp
    gemm(y_hd1, 70, 70 * HWL, nullptr, 0, 0, nullptr, 0, 0, w_hd2, 70, 70, y_hd2, st(9));
    fin(9, bn_hd2, 70);
    act(y_hd2, 9, 70, nullptr, 0, 0, nullptr, 0, 0);
    // xp_new = xp[:,1:7] + xhp[10:70]
    {
        const long total = 2L * 60 * HWL;
        xpnew_kernel<<<(unsigned)((total + 255) / 256), 256, 0, stream>>>(xpnew, xp, y_hd2);
    }
    // pu: [p_fea; xp_new] (316 -> 70)
    gemm(p_fea, 256, 256 * HWL, xpnew, 60, 60 * HWL, nullptr, 0, 0,
         w_pu, 316, 70, y_pu, st(12));
    fin(12, bn_pu, 70);
    // final concat output
    {
        const long total = 2L * 220 * HWL;
        assemble_out<<<(unsigned)((total + 255) / 256), 256, 0, stream>>>(
            (float*)d_out, y_pu, sl(12), y_hu, sl(11), y_fu, sl(10), y_fd2, y_hd2);
    }
}